// MipRayMarcher_76948634075542
// MI455X (gfx1250) — compile-verified
//
#include <hip/hip_runtime.h>
#include <math.h>
#include <stdint.h>

// MipRayMarcher on MI455X (gfx1250): pure streaming kernel (~288MB traffic,
// ~12.4us roofline at 23.3 TB/s). One wave32 per ray; per-ray sample data is
// double-buffered into LDS with CDNA5 async global->LDS DMA (ASYNCcnt path),
// cumprod done as a wave32 shuffle prefix-product. Transcendentals use the
// native base-2 hardware ops (v_exp_f32 / v_log_f32) so VALU issue stays far
// below the HBM roofline; streaming traffic uses NT cache hints.

#define WPB   4          // waves per block (128 threads)
#define RPW   8          // rays processed per wave (pipeline depth)
#define SM    128        // samples per ray
#define SI    127        // intervals per ray
#define EPSV  1e-10f
#define LOG2E 1.4426950408889634f

// per-(wave,buffer) LDS floats: 128 depths + 128 logits + 384 colors + 16 pad
#define BUF_F 656

#define S_WAIT_ASYNC(n) asm volatile("s_wait_asynccnt " #n ::: "memory")

__device__ __forceinline__ void async_ld_b128(uint32_t lds_byte, const float* gsrc) {
    // VDST = LDS byte address (vgpr), VADDR = 64-bit global address (vgpr pair)
    // NT hint: inputs are read exactly once -> don't displace L2 lines.
    asm volatile("global_load_async_to_lds_b128 %0, %1, off th:TH_LOAD_NT"
                 :: "v"(lds_byte), "v"((uint64_t)(uintptr_t)gsrc)
                 : "memory");
}

__device__ __forceinline__ void issue_ray(const float* __restrict__ colors,
                                          const float* __restrict__ dlg,
                                          const float* __restrict__ deps,
                                          long ray, uint32_t ldsBase, int lane) {
    const float* dg = deps   + (size_t)ray * SM       + 4 * lane;
    const float* qg = dlg    + (size_t)ray * SM       + 4 * lane;
    const float* cg = colors + (size_t)ray * (SM * 3) + 4 * lane;
    async_ld_b128(ldsBase +            16u * lane, dg);        // depths  -> buf[0..127]
    async_ld_b128(ldsBase +  512u +    16u * lane, qg);        // logits  -> buf[128..255]
    async_ld_b128(ldsBase + 1024u +    16u * lane, cg);        // colors  -> buf[256..]
    async_ld_b128(ldsBase + 1536u +    16u * lane, cg + 128);
    async_ld_b128(ldsBase + 2048u +    16u * lane, cg + 256);
}

__global__ void __launch_bounds__(WPB * 32)
mip_raymarch_kernel(const float* __restrict__ colors,
                    const float* __restrict__ dlg,
                    const float* __restrict__ deps,
                    float* __restrict__ out,
                    int nrays)
{
    __shared__ __align__(16) float smem[WPB * 2 * BUF_F];

    const int  lane = threadIdx.x & 31;
    const int  wave = threadIdx.x >> 5;
    const long ray0 = ((long)blockIdx.x * WPB + wave) * RPW;
    if (ray0 >= nrays) return;

    // output section offsets (floats), tuple order of the reference
    const size_t nr        = (size_t)nrays;
    const size_t OFF_RGB   = 0;
    const size_t OFF_DEPTH = nr * 3;
    const size_t OFF_W     = nr * 4;
    const size_t OFF_WALL  = OFF_W    + nr * (size_t)SI;
    const size_t OFF_ALPHA = OFF_WALL + nr * (size_t)SI;
    const size_t OFF_DMID  = OFF_ALPHA+ nr * (size_t)SI;

    float* bufp[2] = { &smem[(wave * 2 + 0) * BUF_F], &smem[(wave * 2 + 1) * BUF_F] };
    // low 32 bits of a generic pointer to LDS == the LDS byte address
    uint32_t ldsb[2] = { (uint32_t)(uintptr_t)bufp[0], (uint32_t)(uintptr_t)bufp[1] };

    issue_ray(colors, dlg, deps, ray0, ldsb[0], lane);

    for (int r = 0; r < RPW; ++r) {
        long ray = ray0 + r;
        if (ray >= nrays) break;
        bool havenext = (r + 1 < RPW) && (ray + 1 < nrays);
        if (havenext) {
            issue_ray(colors, dlg, deps, ray + 1, ldsb[(r + 1) & 1], lane);
            S_WAIT_ASYNC(5);   // current ray's 5 DMAs done; next ray's 5 in flight
        } else {
            S_WAIT_ASYNC(0);
        }

        const float* buf = bufp[r & 1];
        const float* dep = buf;
        const float* qv  = buf + SM;
        const float* col = buf + 2 * SM;

        float d[5], x[5], c[15];
        #pragma unroll
        for (int j = 0; j < 5; ++j) { d[j] = dep[4 * lane + j]; x[j] = qv[4 * lane + j]; }
        #pragma unroll
        for (int j = 0; j < 15; ++j) c[j] = col[12 * lane + j];

        float alpha[4], f[4], dmid[4], wv[4], cm0[4], cm1[4], cm2[4];
        #pragma unroll
        for (int k = 0; k < 4; ++k) {
            int   i     = 4 * lane + k;
            float delta = d[k + 1] - d[k];
            float dm    = 0.5f * (d[k + 1] + d[k]);
            float xm    = 0.5f * (x[k + 1] + x[k]) - 1.0f;
            // softplus in base 2: log2(1+e^x) = max(x,0)*log2e + log2(1+2^(-|x|*log2e))
            float l2    = fmaxf(xm, 0.0f) * LOG2E
                        + log2f(1.0f + exp2f(-fabsf(xm) * LOG2E));
            // e^{-softplus*delta} = 2^{-delta * log2(1+e^x)}
            float e     = exp2f(-delta * l2);
            float a     = 1.0f - e;
            float ff    = e + EPSV;                                    // 1 - alpha + eps
            if (i >= SI) { a = 0.0f; ff = 1.0f; dm = 0.0f; }           // lane31 k=3: no interval 127
            alpha[k] = a; f[k] = ff; dmid[k] = dm;
            cm0[k] = 0.5f * (c[3 * k + 0] + c[3 * k + 3]);
            cm1[k] = 0.5f * (c[3 * k + 1] + c[3 * k + 4]);
            cm2[k] = 0.5f * (c[3 * k + 2] + c[3 * k + 5]);
        }

        // wave32 exclusive prefix product of transmittance factors
        float t1 = f[0], t2 = t1 * f[1], t3 = t2 * f[2];
        float incl = t3 * f[3];
        #pragma unroll
        for (int off = 1; off < 32; off <<= 1) {
            float tmp = __shfl_up(incl, off);
            if (lane >= off) incl *= tmp;
        }
        float E = __shfl_up(incl, 1);
        if (lane == 0) E = 1.0f;

        float T[4] = { E, E * t1, E * t2, E * t3 };
        float wsum = 0.f, wd = 0.f, wc0 = 0.f, wc1 = 0.f, wc2 = 0.f;
        #pragma unroll
        for (int k = 0; k < 4; ++k) {
            int   i  = 4 * lane + k;
            float wk = (alpha[k] + EPSV) * T[k];
            if (i >= SI) wk = 0.0f;
            wv[k] = wk;
            wsum += wk;
            wd   += wk * dmid[k];
            wc0  += wk * cm0[k];
            wc1  += wk * cm1[k];
            wc2  += wk * cm2[k];
        }

        // per-interval outputs: streaming, written once -> non-temporal stores
        size_t ob = (size_t)ray * SI;
        #pragma unroll
        for (int k = 0; k < 4; ++k) {
            int i = 4 * lane + k;
            if (i < SI) {
                __builtin_nontemporal_store(wv[k],    &out[OFF_W     + ob + i]);
                __builtin_nontemporal_store(wv[k],    &out[OFF_WALL  + ob + i]);
                __builtin_nontemporal_store(alpha[k], &out[OFF_ALPHA + ob + i]);
                __builtin_nontemporal_store(dmid[k],  &out[OFF_DMID  + ob + i]);
            }
        }

        // wave reduction of 5 scalars
        #pragma unroll
        for (int m = 16; m >= 1; m >>= 1) {
            wsum += __shfl_xor(wsum, m);
            wd   += __shfl_xor(wd, m);
            wc0  += __shfl_xor(wc0, m);
            wc1  += __shfl_xor(wc1, m);
            wc2  += __shfl_xor(wc2, m);
        }
        if (lane == 0) {
            out[OFF_RGB + (size_t)ray * 3 + 0] = wc0 * 2.0f - 1.0f;
            out[OFF_RGB + (size_t)ray * 3 + 1] = wc1 * 2.0f - 1.0f;
            out[OFF_RGB + (size_t)ray * 3 + 2] = wc2 * 2.0f - 1.0f;
            float dpt = wd / (EPSV + wsum);
            if (dpt != dpt) dpt = 100.0f;              // nan_to_num(nan=MAX_DEPTH)
            dpt = fminf(fmaxf(dpt, 0.1f), 100.0f);     // clip(MIN_DEPTH, MAX_DEPTH)
            out[OFF_DEPTH + ray] = dpt;
        }
    }
}

extern "C" void kernel_launch(void* const* d_in, const int* in_sizes, int n_in,
                              void* d_out, int out_size, void* d_ws, size_t ws_size,
                              hipStream_t stream) {
    const float* colors = (const float*)d_in[0];   // [B,R,S,3] f32
    const float* dlg    = (const float*)d_in[1];   // [B,R,S,1] f32
    const float* deps   = (const float*)d_in[2];   // [B,R,S,1] f32
    float* out = (float*)d_out;

    int nrays = in_sizes[1] / SM;                  // B*R (= 65536)
    int raysPerBlock = WPB * RPW;
    int blocks = (nrays + raysPerBlock - 1) / raysPerBlock;
    mip_raymarch_kernel<<<blocks, WPB * 32, 0, stream>>>(colors, dlg, deps, out, nrays);
}